// GCNN_66675072303372
// MI455X (gfx1250) — compile-verified
//
#include <hip/hip_runtime.h>
#include <cstdint>

typedef __attribute__((ext_vector_type(2))) float v2f;
typedef __attribute__((ext_vector_type(8))) float v8f;

#define BATCH 8
#define NPTS  2048
#define TK    20
#define KC    64
#define LPAD  4      // LDS row padding (floats): keeps 16B alignment, breaks bank conflicts
#define SLOPE 0.2f

__device__ __forceinline__ v8f wmma4(v2f a, v2f b, v8f c) {
  // V_WMMA_F32_16X16X4_F32 : D = A(16x4,f32) * B(4x16,f32) + C(16x16,f32)
  return __builtin_amdgcn_wmma_f32_16x16x4_f32(false, a, false, b, (short)0, c, false, false);
}

// --- CDNA5 async global->LDS copy (ASYNCcnt tracked), 16 bytes per lane ------
__device__ __forceinline__ void async_ld_b128(void* lds, const void* g) {
  asm volatile("global_load_async_to_lds_b128 %0, %1, off"
               :: "v"((uint32_t)(uintptr_t)lds), "v"(g) : "memory");
}
__device__ __forceinline__ void wait_async() {
  asm volatile("s_wait_asynccnt 0" ::: "memory");
}

__device__ __forceinline__ bool better(float v1, int j1, float v2, int j2) {
  return (v1 > v2) || (v1 == v2 && j1 < j2);
}

// ---------------- input transpose + pad: x[B,3,N] -> X0[B,N,4] ----------------
__global__ void xin_kernel(const float* __restrict__ x, float* __restrict__ X0) {
  int i = blockIdx.x * blockDim.x + threadIdx.x;
  int total = BATCH * NPTS * 4;
  if (i >= total) return;
  int c = i & 3;
  int n = (i >> 2) % NPTS;
  int b = i / (4 * NPTS);
  X0[i] = (c < 3) ? x[((size_t)b * 3 + c) * NPTS + n] : 0.0f;
}

// ---------------- per-point squared norms: X[B,N,Cpad] -> xx[B,N] -------------
__global__ void norms_kernel(const float* __restrict__ X, float* __restrict__ xx, int Cpad) {
  int i = blockIdx.x * blockDim.x + threadIdx.x;
  if (i >= BATCH * NPTS) return;
  const float* p = X + (size_t)i * Cpad;
  float s = 0.f;
  for (int c = 0; c < Cpad; ++c) s += p[c] * p[c];
  xx[i] = s;
}

// ---------------- W prep ------------------------------------------------------
// split==1: W is [Co, 2*Creal]; Wa = W[:, :Creal] (zero pad to Cpad),
//           Wd = W[:, Creal:] - W[:, :Creal]  (zero pad)
// split==0: W is [Co, Creal]; Wa = pad-copy
__global__ void wprep_kernel(const float* __restrict__ W, float* __restrict__ Wa,
                             float* __restrict__ Wd, int Co, int Creal, int Cpad, int split) {
  int i = blockIdx.x * blockDim.x + threadIdx.x;
  if (i >= Co * Cpad) return;
  int o = i / Cpad, c = i % Cpad;
  if (split) {
    float wa = 0.f, wd = 0.f;
    if (c < Creal) {
      float w1 = W[(size_t)o * (2 * Creal) + c];
      float w2 = W[(size_t)o * (2 * Creal) + Creal + c];
      wa = w1; wd = w2 - w1;
    }
    Wa[i] = wa; Wd[i] = wd;
  } else {
    Wa[i] = (c < Creal) ? W[(size_t)o * Creal + c] : 0.f;
  }
}

// ---------------- fused pairwise-inner (WMMA f32) + streaming top-20 ----------
// grid: (N/64, B), block 128 (4 waves). Wave w owns query rows i0 = 64*bx + 16*w.
// rank value = 2*<x_i,x_j> - ||x_j||^2  (per-row -||x_i||^2 constant dropped)
__global__ void topk_kernel(const float* __restrict__ X, const float* __restrict__ xx,
                            int* __restrict__ idx, int Cpad) {
  extern __shared__ float smem[];
  const int stride = Cpad + LPAD;
  float* sXq = smem;                         // [64][stride]
  float* sXc = sXq + 64 * stride;            // [16][stride]
  float* sS  = sXc + 16 * stride;            // [4][16][16]
  const int b = blockIdx.y;
  const int i0wg = blockIdx.x * 64;
  const int tid = threadIdx.x;
  const int wave = tid >> 5, lane = tid & 31;
  const int l = lane & 15, h = lane >> 4;
  const float* Xb = X + (size_t)b * NPTS * Cpad;
  const float* xxb = xx + (size_t)b * NPTS;
  const int qpr = Cpad >> 2;                 // 16B quads per row

  // stage Xq (64 rows) via async copies
  for (int t = tid; t < 64 * qpr; t += 128) {
    int r = t / qpr, q = t - r * qpr;
    async_ld_b128(&sXq[r * stride + 4 * q], &Xb[(size_t)(i0wg + r) * Cpad + 4 * q]);
  }
  wait_async();

  float tv[TK]; int tj[TK];
#pragma unroll
  for (int t = 0; t < TK; ++t) { tv[t] = -3.4e38f; tj[t] = 0x7fffffff; }

  const float* ap = &sXq[(wave * 16 + l) * stride + 2 * h];
  float* Sw = sS + wave * 256;

  for (int j0 = 0; j0 < NPTS; j0 += 16) {
    __syncthreads();                                  // prev iter done with sXc/sS
    for (int t = tid; t < 16 * qpr; t += 128) {
      int r = t / qpr, q = t - r * qpr;
      async_ld_b128(&sXc[r * stride + 4 * q], &Xb[(size_t)(j0 + r) * Cpad + 4 * q]);
    }
    wait_async();
    __syncthreads();

    v8f acc = {};
    const float* bp = &sXc[l * stride + 2 * h];
    for (int kb = 0; kb < Cpad; kb += 4) {
      v2f a; a[0] = ap[kb]; a[1] = ap[kb + 1];
      v2f bb; bb[0] = bp[kb]; bb[1] = bp[kb + 1];
      acc = wmma4(a, bb, acc);
    }
    float xj = xxb[j0 + l];
#pragma unroll
    for (int v = 0; v < 8; ++v)
      Sw[(v + 8 * h) * 16 + l] = 2.0f * acc[v] - xj;   // S[m][n], col j = j0+n

    if (h == 0) {                                      // lane l scans query row l
      for (int t = 0; t < 16; ++t) {
        float cv = Sw[l * 16 + t];
        int cj = j0 + t;
        if (better(cv, cj, tv[TK - 1], tj[TK - 1])) {
#pragma unroll
          for (int q = 0; q < TK; ++q) {
            bool sw = better(cv, cj, tv[q], tj[q]);
            float ov = tv[q]; int oj = tj[q];
            tv[q] = sw ? cv : ov; tj[q] = sw ? cj : oj;
            cv = sw ? ov : cv;   cj = sw ? oj : cj;
          }
        }
      }
    }
  }
  if (h == 0) {
    int qi = i0wg + wave * 16 + l;
    int* op = idx + ((size_t)b * NPTS + qi) * TK;
#pragma unroll
    for (int t = 0; t < TK; ++t) op[t] = tj[t];
  }
}

// ---------------- point-wise GEMM (WMMA f32): Y[b,n,o] = sum_c W[o,c]*X[b,n,c] -
// grid (N/64, Co/16, B), block 128 (4 waves), K chunked through LDS (async staged).
__global__ void gemm_kernel(const float* __restrict__ X, const float* __restrict__ W,
                            float* __restrict__ Y, int Cpad, int Co) {
  extern __shared__ float smem[];
  const int stride = KC + LPAD;
  float* sX = smem;                 // [64][KC+LPAD]
  float* sW = sX + 64 * stride;     // [16][KC+LPAD]
  const int b = blockIdx.z;
  const int n0 = blockIdx.x * 64;
  const int o0 = blockIdx.y * 16;
  const int tid = threadIdx.x;
  const int wave = tid >> 5, lane = tid & 31;
  const int l = lane & 15, h = lane >> 4;
  const float* Xb = X + (size_t)b * NPTS * Cpad;

  v8f acc = {};
  for (int c0 = 0; c0 < Cpad; c0 += KC) {
    int kc = (Cpad - c0 < KC) ? (Cpad - c0) : KC;
    int qpr = kc >> 2;
    __syncthreads();
    for (int t = tid; t < 64 * qpr; t += 128) {
      int r = t / qpr, q = t - r * qpr;
      async_ld_b128(&sX[r * stride + 4 * q], &Xb[(size_t)(n0 + r) * Cpad + c0 + 4 * q]);
    }
    for (int t = tid; t < 16 * qpr; t += 128) {
      int r = t / qpr, q = t - r * qpr;
      async_ld_b128(&sW[r * stride + 4 * q], &W[(size_t)(o0 + r) * Cpad + c0 + 4 * q]);
    }
    if (c0 + KC < Cpad) __builtin_prefetch(&Xb[(size_t)n0 * Cpad + c0 + KC], 0, 1);
    wait_async();
    __syncthreads();
    const float* ap = &sX[(wave * 16 + l) * stride + 2 * h];
    const float* bp = &sW[l * stride + 2 * h];
    for (int kb = 0; kb < kc; kb += 4) {
      v2f a; a[0] = ap[kb]; a[1] = ap[kb + 1];
      v2f bb; bb[0] = bp[kb]; bb[1] = bp[kb + 1];
      acc = wmma4(a, bb, acc);
    }
  }
  float* Yb = Y + ((size_t)b * NPTS + n0 + wave * 16) * Co;
#pragma unroll
  for (int v = 0; v < 8; ++v)
    Yb[(size_t)(v + 8 * h) * Co + o0 + l] = acc[v];
}

// ---------------- gather + max over k + per-channel partial stats -------------
// grid: B*N/32 blocks, block = Co threads (thread = channel o)
__global__ void gather_max_stats_kernel(const float* __restrict__ U, const float* __restrict__ V,
                                        const int* __restrict__ idx, float* __restrict__ Y,
                                        float* __restrict__ partial, int Co) {
  __shared__ int sidx[32 * TK];
  const int nb = NPTS / 32;
  const int b = blockIdx.x / nb;
  const int n0 = (blockIdx.x % nb) * 32;
  const int o = threadIdx.x;
  for (int t = o; t < 32 * TK; t += Co)
    sidx[t] = idx[((size_t)b * NPTS + n0 + t / TK) * TK + t % TK];
  __syncthreads();
  float s = 0.f, ss = 0.f;
  const float* Ub = U + (size_t)b * NPTS * Co;
  const float* Vb = V + (size_t)b * NPTS * Co;
  for (int p = 0; p < 32; ++p) {
    float vn = Vb[(size_t)(n0 + p) * Co + o];
    float m = -3.4e38f;
#pragma unroll
    for (int k = 0; k < TK; ++k) {
      int j = sidx[p * TK + k];
      float hv = Ub[(size_t)j * Co + o] + vn;
      m = fmaxf(m, hv); s += hv; ss += hv * hv;
    }
    Y[((size_t)b * NPTS + n0 + p) * Co + o] = m;
  }
  partial[(size_t)blockIdx.x * 2 * Co + o] = s;
  partial[(size_t)blockIdx.x * 2 * Co + Co + o] = ss;
}

// ---------------- plain per-channel partial stats over Y[B,N,Co] --------------
__global__ void stats_kernel(const float* __restrict__ Yv, float* __restrict__ partial, int Co) {
  const int nb = NPTS / 32;
  const int b = blockIdx.x / nb;
  const int n0 = (blockIdx.x % nb) * 32;
  const int o = threadIdx.x;
  float s = 0.f, ss = 0.f;
  for (int p = 0; p < 32; ++p) {
    float y = Yv[((size_t)b * NPTS + n0 + p) * Co + o];
    s += y; ss += y * y;
  }
  partial[(size_t)blockIdx.x * 2 * Co + o] = s;
  partial[(size_t)blockIdx.x * 2 * Co + Co + o] = ss;
}

// ---------------- deterministic fixed-order reduction -> scale/shift ----------
__global__ void reduce_stats_kernel(const float* __restrict__ partial, int ngroups, float invcnt,
                                    const float* __restrict__ gamma, const float* __restrict__ beta,
                                    float* __restrict__ scsh, int Co) {
  int o = threadIdx.x;
  float s = 0.f, ss = 0.f;
  for (int g = 0; g < ngroups; ++g) {
    s  += partial[(size_t)g * 2 * Co + o];
    ss += partial[(size_t)g * 2 * Co + Co + o];
  }
  float mean = s * invcnt;
  float var = ss * invcnt - mean * mean;
  float sc = gamma[o] * rsqrtf(var + 1e-5f);
  scsh[o] = sc;
  scsh[Co + o] = beta[o] - mean * sc;
}

// ---------------- BN + LeakyReLU in place on Y[B,N,Co] ------------------------
__global__ void bn_lrelu_kernel(float* __restrict__ Y, const float* __restrict__ scsh, int Co) {
  int i = blockIdx.x * blockDim.x + threadIdx.x;
  if (i >= BATCH * NPTS * Co) return;
  int o = i % Co;
  float y = Y[i] * scsh[o] + scsh[Co + o];
  Y[i] = (y >= 0.f) ? y : SLOPE * y;
}

// ---------------- BN + LeakyReLU + transpose to d_out[B,Co,N] -----------------
__global__ void bn_lrelu_tr_kernel(const float* __restrict__ H, const float* __restrict__ scsh,
                                   float* __restrict__ out, int Co) {
  int i = blockIdx.x * blockDim.x + threadIdx.x;
  if (i >= BATCH * NPTS * Co) return;
  int o = i % Co;
  int n = (i / Co) % NPTS;
  int b = i / (Co * NPTS);
  float y = H[i] * scsh[o] + scsh[Co + o];
  y = (y >= 0.f) ? y : SLOPE * y;
  out[((size_t)b * Co + o) * NPTS + n] = y;
}

// ---------------- concat X1..X4 -> XC[B,N,512] --------------------------------
__global__ void concat_kernel(const float* __restrict__ X1, const float* __restrict__ X2,
                              const float* __restrict__ X3, const float* __restrict__ X4,
                              float* __restrict__ XC) {
  int i = blockIdx.x * blockDim.x + threadIdx.x;
  if (i >= BATCH * NPTS * 512) return;
  int c = i % 512;
  size_t bn = (size_t)(i / 512);
  float v;
  if (c < 64)       v = X1[bn * 64 + c];
  else if (c < 128) v = X2[bn * 64 + (c - 64)];
  else if (c < 256) v = X3[bn * 128 + (c - 128)];
  else              v = X4[bn * 256 + (c - 256)];
  XC[i] = v;
}

extern "C" void kernel_launch(void* const* d_in, const int* in_sizes, int n_in,
                              void* d_out, int out_size, void* d_ws, size_t ws_size,
                              hipStream_t stream) {
  const size_t BN = (size_t)BATCH * NPTS;
  float* ws = (float*)d_ws;
  size_t off = 0;
  auto alloc = [&](size_t n) { float* p = ws + off; off += n; return p; };

  float* X0 = alloc(BN * 4);
  float* X1 = alloc(BN * 64);
  float* X2 = alloc(BN * 64);
  float* X3 = alloc(BN * 128);
  float* X4 = alloc(BN * 256);
  float* U  = alloc(BN * 256);        // U and V are contiguous:
  float* V  = alloc(BN * 256);        // after block 4, [U|V] is reused as XC[B,N,512]
  float* H5 = alloc(BN * 128);
  float* xx = alloc(BN);
  float* Wa = alloc(128 * 512);       // >= 256*256 and == 128*512
  float* Wd = alloc(128 * 512);
  int*   idx = (int*)alloc(BN * TK);
  float* partial = alloc((BN / 32) * 2 * 256);
  float* scsh = alloc(2 * 256);
  float* XC = U;

  const int ngroups = (int)(BN / 32);
  const size_t gemm_shmem = (size_t)(64 + 16) * (KC + LPAD) * sizeof(float);

  struct Cfg { const float *Win, *g, *bt; float *Xin, *Xout; int Creal, Cpad, Co; };
  Cfg cfg[4] = {
    { (const float*)d_in[1],  (const float*)d_in[2],  (const float*)d_in[3],  X0, X1,   3,   4,  64 },
    { (const float*)d_in[4],  (const float*)d_in[5],  (const float*)d_in[6],  X1, X2,  64,  64,  64 },
    { (const float*)d_in[7],  (const float*)d_in[8],  (const float*)d_in[9],  X2, X3,  64,  64, 128 },
    { (const float*)d_in[10], (const float*)d_in[11], (const float*)d_in[12], X3, X4, 128, 128, 256 },
  };

  // input transpose + pad
  {
    int tot = BATCH * NPTS * 4;
    xin_kernel<<<(tot + 255) / 256, 256, 0, stream>>>((const float*)d_in[0], X0);
  }

  for (int blk = 0; blk < 4; ++blk) {
    const Cfg& c = cfg[blk];

    norms_kernel<<<(int)((BN + 255) / 256), 256, 0, stream>>>(c.Xin, xx, c.Cpad);

    size_t topk_shmem = ((size_t)(64 + 16) * (c.Cpad + LPAD) + 4 * 16 * 16) * sizeof(float);
    topk_kernel<<<dim3(NPTS / 64, BATCH), 128, topk_shmem, stream>>>(c.Xin, xx, idx, c.Cpad);

    int wtot = c.Co * c.Cpad;
    wprep_kernel<<<(wtot + 255) / 256, 256, 0, stream>>>(c.Win, Wa, Wd, c.Co, c.Creal, c.Cpad, 1);

    dim3 ggrid(NPTS / 64, c.Co / 16, BATCH);
    gemm_kernel<<<ggrid, 128, gemm_shmem, stream>>>(c.Xin, Wa, U, c.Cpad, c.Co);
    gemm_kernel<<<ggrid, 128, gemm_shmem, stream>>>(c.Xin, Wd, V, c.Cpad, c.Co);

    gather_max_stats_kernel<<<ngroups, c.Co, 0, stream>>>(U, V, idx, c.Xout, partial, c.Co);

    float invcnt = 1.0f / (float)(BN * TK);
    reduce_stats_kernel<<<1, c.Co, 0, stream>>>(partial, ngroups, invcnt, c.g, c.bt, scsh, c.Co);

    int tot = (int)(BN * c.Co);
    bn_lrelu_kernel<<<(tot + 255) / 256, 256, 0, stream>>>(c.Xout, scsh, c.Co);
  }

  // concat -> XC (aliases U|V region, safe: U/V dead now)
  {
    int tot = (int)(BN * 512);
    concat_kernel<<<(tot + 255) / 256, 256, 0, stream>>>(X1, X2, X3, X4, XC);
  }

  // final 1x1 conv: W5[128,512] (no split), BN over (0,2), LReLU, transpose out
  {
    const float* W5 = (const float*)d_in[13];
    const float* g5 = (const float*)d_in[14];
    const float* b5 = (const float*)d_in[15];
    wprep_kernel<<<(128 * 512 + 255) / 256, 256, 0, stream>>>(W5, Wa, Wd, 128, 512, 512, 0);

    dim3 ggrid(NPTS / 64, 128 / 16, BATCH);
    gemm_kernel<<<ggrid, 128, gemm_shmem, stream>>>(XC, Wa, H5, 512, 128);

    stats_kernel<<<ngroups, 128, 0, stream>>>(H5, partial, 128);
    float invcnt = 1.0f / (float)BN;
    reduce_stats_kernel<<<1, 128, 0, stream>>>(partial, ngroups, invcnt, g5, b5, scsh, 128);

    int tot = (int)(BN * 128);
    bn_lrelu_tr_kernel<<<(tot + 255) / 256, 256, 0, stream>>>(H5, scsh, (float*)d_out, 128);
  }
}